// SpectralConv1d_45500883534283
// MI455X (gfx1250) — compile-verified
//
#include <hip/hip_runtime.h>
#include <hip/hip_bf16.h>

typedef __attribute__((ext_vector_type(2))) float v2f;
typedef __attribute__((ext_vector_type(4))) float v4f;
typedef __attribute__((ext_vector_type(8))) float v8f;

#define W_LEN   4096
#define IN_CH   128
#define D_VARS  128
#define N_MODES 4096
#define NPAD_F  8191   // forward rfft length (2W-1)
#define NPAD_I  8190   // irfft length 2*(K-1)
#define TWO_PI  6.28318530717958647692f

// ---------------------------------------------------------------------------
// Kernel A: forward DFT as GEMM, X[k,i] = (1/sqrt(8191)) * sum_w x[w,i]*e^{-j2pi kw/8191}
// One wave per 16(k) x 16(i) tile; A = on-the-fly twiddles (16x4), B = x tile (4x16).
// Output stored channel-major: Xr[i*4096 + k] so the streaming kernel reads it
// contiguously in k (L2-resident, 2 MB each).
// ---------------------------------------------------------------------------
__global__ __launch_bounds__(256) void fwd_dft(const float* __restrict__ x,
                                               float* __restrict__ Xr,
                                               float* __restrict__ Xi) {
  const int lane = threadIdx.x & 31;
  const int wave = threadIdx.x >> 5;
  const int k0   = blockIdx.x * 16;   // mode tile
  const int i0   = wave * 16;        // channel tile (8 waves cover 128)
  const int row  = lane & 15;        // M index within tile
  const int kh   = lane >> 4;        // half-select for A/B K striping
  const int col  = lane & 15;        // N index within tile
  const int kidx = k0 + row;
  const float wstep = -TWO_PI / (float)NPAD_F;

  v8f cr = {};
  v8f ci = {};
  for (int w0 = 0; w0 < W_LEN; w0 += 4) {
    v2f ac, as, b;
#pragma unroll
    for (int t = 0; t < 2; ++t) {
      // exact integer phase reduction: k*w <= 4095^2 < 2^24, then mod 8191
      int kw = (kidx * (w0 + kh * 2 + t)) % NPAD_F;
      float ang = wstep * (float)kw;
      ac[t] = __cosf(ang);
      as[t] = __sinf(ang);   // negative angle => -sin folded in (imag part)
    }
    // B 4x16 tile of x: VGPR t holds K = 2t + kh (lanes striped by kh)
    b[0] = x[(w0 + kh) * IN_CH + i0 + col];
    b[1] = x[(w0 + 2 + kh) * IN_CH + i0 + col];
    cr = __builtin_amdgcn_wmma_f32_16x16x4_f32(false, ac, false, b, (short)0, cr, false, false);
    ci = __builtin_amdgcn_wmma_f32_16x16x4_f32(false, as, false, b, (short)0, ci, false, false);
  }

  const float fs = rsqrtf((float)NPAD_F);   // ortho norm
#pragma unroll
  for (int v = 0; v < 8; ++v) {
    int m = v + 8 * kh;  // k within tile (C/D layout: lanes>=16 hold M+8)
    Xr[(size_t)(i0 + col) * N_MODES + (k0 + m)] = cr[v] * fs;
    Xi[(size_t)(i0 + col) * N_MODES + (k0 + m)] = ci[v] * fs;
  }
}

// ---------------------------------------------------------------------------
// Kernel B: the bandwidth floor. Xk[k,o] = sum_i X[k,i] * (Wr + jWi)[i,o,k].
// Weight is contiguous in k for fixed (i,o): each thread owns 4 consecutive k
// for one o and streams the i loop with b128 loads -> every weight byte read
// exactly once, fully coalesced (537 MB / 23.3 TB/s ~ 23 us).
// irfft edge/2x ortho scaling is folded into the stores. Output stored
// o-major (contiguous v4 store), consumed from L2 by the inverse DFT.
// ---------------------------------------------------------------------------
__global__ __launch_bounds__(256) void mode_mix(const float* __restrict__ Wr,
                                                const float* __restrict__ Wi,
                                                const float* __restrict__ Xr,
                                                const float* __restrict__ Xi,
                                                float* __restrict__ Ykr,
                                                float* __restrict__ Yki) {
  const int o = blockIdx.y;
  const int k = (blockIdx.x * 256 + (int)threadIdx.x) * 4;
  const size_t DN = (size_t)D_VARS * N_MODES;   // i-stride in weights

  const float* wrp = Wr + (size_t)o * N_MODES + k;
  const float* wip = Wi + (size_t)o * N_MODES + k;

  v4f ar = {};
  v4f ai = {};
  for (int i = 0; i < IN_CH; ++i) {
    v4f wr = *(const v4f*)(wrp + (size_t)i * DN);
    v4f wi = *(const v4f*)(wip + (size_t)i * DN);
    v4f xr = *(const v4f*)(Xr + (size_t)i * N_MODES + k);
    v4f xi = *(const v4f*)(Xi + (size_t)i * N_MODES + k);
    ar += xr * wr - xi * wi;
    ai += xr * wi + xi * wr;
    int ip = (i + 4 < IN_CH) ? i + 4 : IN_CH - 1;
    __builtin_prefetch(wrp + (size_t)ip * DN, 0, 0);
    __builtin_prefetch(wip + (size_t)ip * DN, 0, 0);
  }

  const float is = rsqrtf((float)NPAD_I);   // irfft ortho norm
  v4f sr, si;
#pragma unroll
  for (int j = 0; j < 4; ++j) {
    int kk = k + j;
    float s = (kk == 0 || kk == N_MODES - 1) ? is : 2.0f * is;  // hermitian fold
    sr[j] = ar[j] * s;
    si[j] = ai[j] * s;
  }
  *(v4f*)(Ykr + (size_t)o * N_MODES + k) = sr;
  *(v4f*)(Yki + (size_t)o * N_MODES + k) = si;
}

// ---------------------------------------------------------------------------
// Kernel C: inverse DFT as GEMM, y[w,o] = sum_k s_k*(Xkr*cos(th) - Xki*sin(th)),
// th = 2pi*k*w/8190. Two chained WMMAs per K-step accumulate into one tile.
// ---------------------------------------------------------------------------
__global__ __launch_bounds__(256) void inv_dft(const float* __restrict__ Ykr,
                                               const float* __restrict__ Yki,
                                               float* __restrict__ out) {
  const int lane = threadIdx.x & 31;
  const int wave = threadIdx.x >> 5;
  const int w0   = blockIdx.x * 16;
  const int o0   = wave * 16;
  const int row  = lane & 15;
  const int kh   = lane >> 4;
  const int col  = lane & 15;
  const int wrow = w0 + row;
  const float astep = TWO_PI / (float)NPAD_I;

  v8f acc = {};
  for (int k0 = 0; k0 < N_MODES; k0 += 4) {
    v2f ac, ans, br, bi;
#pragma unroll
    for (int t = 0; t < 2; ++t) {
      int kk = k0 + kh * 2 + t;
      int km = (kk * wrow) % NPAD_I;         // <= 4095*4095 < 2^24, exact
      float ang = astep * (float)km;
      ac[t]  =  __cosf(ang);
      ans[t] = -__sinf(ang);
    }
    br[0] = Ykr[(size_t)(o0 + col) * N_MODES + (k0 + kh)];
    br[1] = Ykr[(size_t)(o0 + col) * N_MODES + (k0 + 2 + kh)];
    bi[0] = Yki[(size_t)(o0 + col) * N_MODES + (k0 + kh)];
    bi[1] = Yki[(size_t)(o0 + col) * N_MODES + (k0 + 2 + kh)];
    acc = __builtin_amdgcn_wmma_f32_16x16x4_f32(false, ac,  false, br, (short)0, acc, false, false);
    acc = __builtin_amdgcn_wmma_f32_16x16x4_f32(false, ans, false, bi, (short)0, acc, false, false);
  }

#pragma unroll
  for (int v = 0; v < 8; ++v) {
    int m = v + 8 * kh;
    out[(size_t)(w0 + m) * D_VARS + (o0 + col)] = acc[v];
  }
}

// ---------------------------------------------------------------------------
extern "C" void kernel_launch(void* const* d_in, const int* in_sizes, int n_in,
                              void* d_out, int out_size, void* d_ws, size_t ws_size,
                              hipStream_t stream) {
  (void)in_sizes; (void)n_in; (void)out_size; (void)ws_size;
  const float* x  = (const float*)d_in[0];
  const float* Wr = (const float*)d_in[1];
  const float* Wi = (const float*)d_in[2];

  // workspace layout: Xr, Xi (channel-major), Ykr, Yki (o-major) : 8 MB total
  float* ws  = (float*)d_ws;
  float* Xr  = ws;
  float* Xi  = Xr  + (size_t)IN_CH  * N_MODES;
  float* Ykr = Xi  + (size_t)IN_CH  * N_MODES;
  float* Yki = Ykr + (size_t)D_VARS * N_MODES;

  fwd_dft <<<dim3(N_MODES / 16),              256, 0, stream>>>(x, Xr, Xi);
  mode_mix<<<dim3(N_MODES / 1024, D_VARS),    256, 0, stream>>>(Wr, Wi, Xr, Xi, Ykr, Yki);
  inv_dft <<<dim3(W_LEN / 16),                256, 0, stream>>>(Ykr, Yki, (float*)d_out);
}